// GraphConv_72868415144453
// MI455X (gfx1250) — compile-verified
//
#include <hip/hip_runtime.h>

typedef float v2f __attribute__((ext_vector_type(2)));
typedef float v8f __attribute__((ext_vector_type(8)));

#define IN_FEATS 128

// ---------------------------------------------------------------------------
// Kernel 1: h = feat @ W^T + b   via V_WMMA_F32_16X16X4_F32
// One wave32 computes one 16(node) x 16(outfeat) tile, K=128 in steps of 4.
// A (16x4 f32) layout: lanes 0-15 M=0..15 {K=k0,k0+1}, lanes 16-31 {K=k0+2,k0+3}
// B (4x16 f32) layout: lanes 0-15 N=0..15 {K=k0,k0+1}, lanes 16-31 {K=k0+2,k0+3}
// C/D (16x16 f32): VGPR r = row (r + 8*(lane>>4)), col = lane&15
//
// Out-of-range rows/cols are CLAMPED (not predicated) on the load side: row m
// of A only feeds row m of D, so clamped lanes compute garbage rows that are
// never stored. EXEC stays all-1s through the whole WMMA loop.
// Store side: full tiles (wave-uniform test) store branch-free; only a tail
// tile pays per-lane guards.
// ---------------------------------------------------------------------------
__global__ void gc_proj_wmma(const float* __restrict__ feat,
                             const float* __restrict__ Ww,
                             const float* __restrict__ Wb,
                             float* __restrict__ h,
                             int n_nodes, int out_feats, int total_tiles) {
    const int wave = threadIdx.x >> 5;
    const int tile = blockIdx.x * (blockDim.x >> 5) + wave;
    if (tile >= total_tiles) return;            // uniform per-wave exit

    const int lane    = threadIdx.x & 31;
    const int ntilesN = out_feats >> 4;
    const int mt = tile / ntilesN;
    const int nt = tile - mt * ntilesN;
    const int row0 = mt << 4;
    const int col0 = nt << 4;

    const int half = lane >> 4;                 // 0: K=k0..k0+1, 1: K=k0+2..k0+3
    const int l15  = lane & 15;
    const int arow = min(row0 + l15, n_nodes - 1);   // clamped: always loadable
    const int bcol = min(col0 + l15, out_feats - 1);

    // bias depends only on N (= l15) -> same across all 8 C registers
    const float bias = Wb[bcol];
    v8f c;
#pragma unroll
    for (int i = 0; i < 8; ++i) c[i] = bias;

    const float2* __restrict__ fa =
        (const float2*)(feat + (size_t)arow * IN_FEATS) + half;
    const float2* __restrict__ fb =
        (const float2*)(Ww   + (size_t)bcol * IN_FEATS) + half;

#pragma unroll 4
    for (int k0 = 0; k0 < IN_FEATS; k0 += 4) {
        const float2 av = fa[k0 >> 1];          // K = k0 + 2*half .. +1
        const float2 bv = fb[k0 >> 1];
        v2f a; a[0] = av.x; a[1] = av.y;
        v2f b; b[0] = bv.x; b[1] = bv.y;
        // 8 args: (neg_a, A, neg_b, B, c_mod, C, reuse_a, reuse_b)
        c = __builtin_amdgcn_wmma_f32_16x16x4_f32(false, a, false, b,
                                                  (short)0, c, false, false);
    }

    // Lane's first output element: row (row0 + 8*half), col (col0 + l15).
    // Successive C registers step the row by 1 -> fixed out_feats stride.
    float* __restrict__ hp =
        h + (size_t)(row0 + (half << 3)) * out_feats + (col0 + l15);

    if (row0 + 16 <= n_nodes) {                 // wave-uniform: full tile
#pragma unroll
        for (int r = 0; r < 8; ++r)
            hp[(size_t)r * out_feats] = c[r];
    } else {                                    // tail tile only
#pragma unroll
        for (int r = 0; r < 8; ++r)
            if (row0 + r + (half << 3) < n_nodes)
                hp[(size_t)r * out_feats] = c[r];
    }
}

// ---------------------------------------------------------------------------
// Kernel 2: scatter  out[dst[e]] += h[src[e]] * e[e]
// One wave per edge: 32 lanes x float2 = 64 feats, one 256B coalesced row.
// h (25.6MB) and out (25.6MB) both resident in the 192MB L2 -> gather hits L2
// and the f32 atomics resolve in L2's atomic units.
// ---------------------------------------------------------------------------
__global__ void gc_edge_scatter(const float* __restrict__ h,
                                const long long* __restrict__ src,
                                const long long* __restrict__ dst,
                                const float* __restrict__ ew,
                                float* __restrict__ out,
                                long long n_edges, int out_feats) {
    const int lane = threadIdx.x & 31;
    const long long wave_in_grid =
        (long long)blockIdx.x * (blockDim.x >> 5) + (threadIdx.x >> 5);
    const long long stride = (long long)gridDim.x * (blockDim.x >> 5);

    for (long long edge = wave_in_grid; edge < n_edges; edge += stride) {
        const long long s = src[edge];
        const long long d = dst[edge];
        const float     w = ew[edge];
        const float2 hv = *(const float2*)(h + s * out_feats + 2 * lane);
        float* o = out + d * out_feats + 2 * lane;
        unsafeAtomicAdd(o,     hv.x * w);   // -> global_atomic_add_f32
        unsafeAtomicAdd(o + 1, hv.y * w);
    }
}

// ---------------------------------------------------------------------------
// Kernel 3/0: zero-init and in-place ReLU on d_out
// ---------------------------------------------------------------------------
__global__ void gc_zero(float* __restrict__ p, int n) {
    int i = blockIdx.x * blockDim.x + threadIdx.x;
    if (i < n) p[i] = 0.0f;
}

__global__ void gc_relu(float* __restrict__ p, int n) {
    int i = blockIdx.x * blockDim.x + threadIdx.x;
    if (i < n) p[i] = fmaxf(p[i], 0.0f);
}

extern "C" void kernel_launch(void* const* d_in, const int* in_sizes, int n_in,
                              void* d_out, int out_size, void* d_ws, size_t ws_size,
                              hipStream_t stream) {
    const float*     feat = (const float*)d_in[0];
    const long long* src  = (const long long*)d_in[1];
    const long long* dst  = (const long long*)d_in[2];
    const float*     ew   = (const float*)d_in[3];
    const float*     Ww   = (const float*)d_in[4];
    const float*     Wb   = (const float*)d_in[5];
    float* out = (float*)d_out;
    float* h   = (float*)d_ws;                  // 100000*64*4 = 25.6 MB scratch

    const int  out_feats = in_sizes[5];         // 64
    const int  n_nodes   = in_sizes[0] / IN_FEATS;
    const long long n_edges = in_sizes[1];
    const int  total = n_nodes * out_feats;     // == out_size

    // 0) zero the accumulator (d_out is poisoned by the harness)
    gc_zero<<<(total + 255) / 256, 256, 0, stream>>>(out, total);

    // 1) dense projection with f32 WMMA into workspace
    const int mtiles = (n_nodes + 15) / 16;
    const int ntiles = out_feats / 16;
    const int tiles  = mtiles * ntiles;         // 6250 * 4 = 25000 waves
    gc_proj_wmma<<<(tiles + 3) / 4, 128, 0, stream>>>(feat, Ww, Wb, h,
                                                      n_nodes, out_feats, tiles);

    // 2) gather * e, atomic scatter-add into out (L2-resident)
    const int wpb = 256 / 32;                   // 8 edges per block
    long long blocks = (n_edges + wpb - 1) / wpb;
    if (blocks > 262144) blocks = 262144;       // grid-stride covers the rest
    gc_edge_scatter<<<(int)blocks, 256, 0, stream>>>(h, src, dst, ew, out,
                                                     n_edges, out_feats);

    // 3) ReLU in place
    gc_relu<<<(total + 255) / 256, 256, 0, stream>>>(out, total);
}